// EnvelopeWassersteinLoss_49151605735882
// MI455X (gfx1250) — compile-verified
//
#include <hip/hip_runtime.h>

typedef float v2f __attribute__((ext_vector_type(2)));
typedef float v8f __attribute__((ext_vector_type(8)));

#define N 8192
#define D 64
#define EPS_REG 0.05f
#define NUM_ITER 20
#define LOSS_BLOCKS 2048

// ---------------------------------------------------------------------------
// Kernel 1: row norms of XP/XQ, init u = v = 1/N, zero the max-C scalar.
// ---------------------------------------------------------------------------
__global__ void __launch_bounds__(256) norms_init_kernel(
    const float* __restrict__ XP, const float* __restrict__ XQ,
    float* __restrict__ np, float* __restrict__ nq,
    float* __restrict__ u, float* __restrict__ v,
    int* __restrict__ cmaxBits)
{
    int i = blockIdx.x * 256 + threadIdx.x;          // 0 .. 2N-1
    if (i == 0) *cmaxBits = 0;                        // float 0.0 bit pattern
    const float* src = (i < N) ? (XP + (size_t)i * D)
                               : (XQ + (size_t)(i - N) * D);
    float s = 0.0f;
#pragma unroll
    for (int k = 0; k < D; ++k) { float x = src[k]; s = fmaf(x, x, s); }
    if (i < N) { np[i] = s; u[i] = 1.0f / (float)N; }
    else       { nq[i - N] = s; v[i - N] = 1.0f / (float)N; }
}

// ---------------------------------------------------------------------------
// Kernel 2: C = relu(np[i] + nq[j] - 2*XP@XQ^T) via f32 WMMA (16x16x4, K=64).
// One wave per 16x16 tile; EXEC all-ones everywhere (WMMA requirement).
// Also reduces max(C) into cmaxBits with an order-independent int atomicMax.
// ---------------------------------------------------------------------------
__global__ void __launch_bounds__(256) cost_gemm_kernel(
    const float* __restrict__ XP, const float* __restrict__ XQ,
    const float* __restrict__ np, const float* __restrict__ nq,
    float* __restrict__ C, int* __restrict__ cmaxBits)
{
    const int lane = threadIdx.x & 31;
    const int wave = threadIdx.x >> 5;
    const int t    = blockIdx.x * 8 + wave;          // tile id, 512x512 tiles
    const int tm   = t >> 9;
    const int tn   = t & 511;
    const int hl   = lane >> 4;                      // lane half: 0 or 1
    const int lm   = lane & 15;

    // A-matrix f32 16x4 layout: lanes 0-15 hold M=lane {K0,K1}; lanes 16-31
    // hold M=lane-16 {K2,K3}.  B mirrors it with N in place of M.
    const float* ap = XP + ((size_t)(tm * 16 + lm)) * D + 2 * hl;
    const float* bp = XQ + ((size_t)(tn * 16 + lm)) * D + 2 * hl;

    v8f acc = {};
#pragma unroll
    for (int kk = 0; kk < 16; ++kk) {                // 16 chunks of K=4
        v2f a = *(const v2f*)(ap + kk * 4);
        v2f b = *(const v2f*)(bp + kk * 4);
        acc = __builtin_amdgcn_wmma_f32_16x16x4_f32(
            /*neg_a=*/false, a, /*neg_b=*/false, b,
            /*c_mod=*/(short)0, acc, /*reuse_a=*/false, /*reuse_b=*/false);
    }

    // C/D layout: lane<16 -> N=lane, VGPR r -> M=r ; lane>=16 -> M=8+r.
    const int col     = tn * 16 + lm;
    const int rowbase = tm * 16 + 8 * hl;
    const float nqj   = nq[col];
    float fmax = 0.0f;
#pragma unroll
    for (int r = 0; r < 8; ++r) {
        float c = np[rowbase + r] + nqj - 2.0f * acc[r];
        c = fmaxf(c, 0.0f);
        C[(size_t)(rowbase + r) * N + col] = c;
        fmax = fmaxf(fmax, c);
    }
    // wave32 butterfly max reduce
    for (int off = 16; off > 0; off >>= 1)
        fmax = fmaxf(fmax, __shfl_xor(fmax, off, 32));
    if (lane == 0) atomicMax(cmaxBits, __float_as_int(fmax)); // C>=0 => valid
}

// ---------------------------------------------------------------------------
// Kernel 3: reg = EPS * max(C); store reg and 1/reg.
// ---------------------------------------------------------------------------
__global__ void reg_kernel(const int* __restrict__ cmaxBits,
                           float* __restrict__ regbuf)
{
    float cmax = __int_as_float(*cmaxBits);
    float reg  = EPS_REG * cmax;
    regbuf[0] = reg;
    regbuf[1] = 1.0f / reg;
}

// ---------------------------------------------------------------------------
// Kernel 4: K = exp(-C/reg) in place (grid-stride over 67M elements).
// ---------------------------------------------------------------------------
__global__ void __launch_bounds__(256) exp_kernel(
    float* __restrict__ K, const float* __restrict__ regbuf)
{
    const float ir = regbuf[1];
    const size_t total  = (size_t)N * N;
    size_t idx    = (size_t)blockIdx.x * 256 + threadIdx.x;
    const size_t stride = (size_t)gridDim.x * 256;
    for (; idx < total; idx += stride)
        K[idx] = __expf(-K[idx] * ir);
}

// ---------------------------------------------------------------------------
// Kernel 5a: partial column sums of K^T u.  grid (32, 64): blockIdx.x covers
// 256 columns, blockIdx.y a 128-row chunk.  Coalesced; deterministic.
// ---------------------------------------------------------------------------
__global__ void __launch_bounds__(256) colsum_partial_kernel(
    const float* __restrict__ K, const float* __restrict__ u,
    float* __restrict__ partial)
{
    const int j  = blockIdx.x * 256 + threadIdx.x;
    const int i0 = blockIdx.y * 128;
    const float* kp = K + (size_t)i0 * N + j;
    float s = 0.0f;
#pragma unroll 4
    for (int r = 0; r < 128; ++r)
        s = fmaf(kp[(size_t)r * N], u[i0 + r], s);
    partial[(size_t)blockIdx.y * N + j] = s;
}

// ---------------------------------------------------------------------------
// Kernel 5b: v[j] = b / sum_c partial[c][j]   (fixed-order reduce over 64)
// ---------------------------------------------------------------------------
__global__ void __launch_bounds__(256) v_update_kernel(
    const float* __restrict__ partial, float* __restrict__ v)
{
    const int j = blockIdx.x * 256 + threadIdx.x;
    float s = 0.0f;
#pragma unroll
    for (int c = 0; c < 64; ++c)
        s += partial[(size_t)c * N + j];
    v[j] = (1.0f / (float)N) / s;
}

// ---------------------------------------------------------------------------
// Kernel 5c: u[i] = a / (K v)[i].  One block per row, coalesced row scan,
// deterministic LDS tree reduce.
// ---------------------------------------------------------------------------
__global__ void __launch_bounds__(256) u_update_kernel(
    const float* __restrict__ K, const float* __restrict__ v,
    float* __restrict__ u)
{
    __shared__ float red[256];
    const float* kp = K + (size_t)blockIdx.x * N;
    float s = 0.0f;
    for (int j = threadIdx.x; j < N; j += 256)
        s = fmaf(kp[j], v[j], s);
    red[threadIdx.x] = s;
    __syncthreads();
    for (int w = 128; w > 0; w >>= 1) {
        if (threadIdx.x < w) red[threadIdx.x] += red[threadIdx.x + w];
        __syncthreads();
    }
    if (threadIdx.x == 0) u[blockIdx.x] = (1.0f / (float)N) / red[0];
}

// ---------------------------------------------------------------------------
// Kernel 6: loss partials: sum_ij  u_i * K_ij * v_j * (-reg * log K_ij)
// (recovers C = -reg*log K instead of storing a second 268MB matrix)
// ---------------------------------------------------------------------------
__global__ void __launch_bounds__(256) loss_partial_kernel(
    const float* __restrict__ K, const float* __restrict__ u,
    const float* __restrict__ v, const float* __restrict__ regbuf,
    float* __restrict__ lossP)
{
    __shared__ float red[256];
    const float reg = regbuf[0];
    const size_t total  = (size_t)N * N;
    size_t idx    = (size_t)blockIdx.x * 256 + threadIdx.x;
    const size_t stride = (size_t)LOSS_BLOCKS * 256;
    float s = 0.0f;
    for (; idx < total; idx += stride) {
        float k = K[idx];
        size_t i = idx >> 13;            // /8192
        size_t j = idx & (N - 1);
        float c = -reg * __logf(k);      // == C_ij
        s = fmaf(u[i] * k * v[j], c, s);
    }
    red[threadIdx.x] = s;
    __syncthreads();
    for (int w = 128; w > 0; w >>= 1) {
        if (threadIdx.x < w) red[threadIdx.x] += red[threadIdx.x + w];
        __syncthreads();
    }
    if (threadIdx.x == 0) lossP[blockIdx.x] = red[0];
}

__global__ void __launch_bounds__(256) loss_final_kernel(
    const float* __restrict__ lossP, float* __restrict__ out)
{
    __shared__ float red[256];
    float s = 0.0f;
    for (int i = threadIdx.x; i < LOSS_BLOCKS; i += 256) s += lossP[i];
    red[threadIdx.x] = s;
    __syncthreads();
    for (int w = 128; w > 0; w >>= 1) {
        if (threadIdx.x < w) red[threadIdx.x] += red[threadIdx.x + w];
        __syncthreads();
    }
    if (threadIdx.x == 0) out[0] = red[0];
}

// ---------------------------------------------------------------------------
extern "C" void kernel_launch(void* const* d_in, const int* in_sizes, int n_in,
                              void* d_out, int out_size, void* d_ws, size_t ws_size,
                              hipStream_t stream)
{
    const float* XP = (const float*)d_in[0];   // 8192 x 64 f32
    const float* XQ = (const float*)d_in[1];   // 8192 x 64 f32
    float* out = (float*)d_out;

    // workspace layout (floats)
    float* ws      = (float*)d_ws;
    float* Kbuf    = ws;                               // N*N      = 67108864
    float* np      = Kbuf + (size_t)N * N;             // N
    float* nq      = np + N;                           // N
    float* u       = nq + N;                           // N
    float* v       = u + N;                            // N
    float* partial = v + N;                            // 64*N     = 524288
    float* lossP   = partial + (size_t)64 * N;         // LOSS_BLOCKS
    float* regbuf  = lossP + LOSS_BLOCKS;              // [0]=reg [1]=1/reg
    int*   cmaxB   = (int*)(regbuf + 4);

    // 1) norms + u/v init + cmax reset
    norms_init_kernel<<<(2 * N) / 256, 256, 0, stream>>>(XP, XQ, np, nq, u, v, cmaxB);
    // 2) cost matrix via f32 WMMA + max reduction
    cost_gemm_kernel<<<(512 * 512) / 8, 256, 0, stream>>>(XP, XQ, np, nq, Kbuf, cmaxB);
    // 3) reg scalar
    reg_kernel<<<1, 1, 0, stream>>>(cmaxB, regbuf);
    // 4) K = exp(-C/reg) in place
    exp_kernel<<<8192, 256, 0, stream>>>(Kbuf, regbuf);
    // 5) Sinkhorn iterations
    for (int it = 0; it < NUM_ITER; ++it) {
        colsum_partial_kernel<<<dim3(N / 256, 64), 256, 0, stream>>>(Kbuf, u, partial);
        v_update_kernel<<<N / 256, 256, 0, stream>>>(partial, v);
        u_update_kernel<<<N, 256, 0, stream>>>(Kbuf, v, u);
    }
    // 6) loss = sum C .* (u K v)
    loss_partial_kernel<<<LOSS_BLOCKS, 256, 0, stream>>>(Kbuf, u, v, regbuf, lossP);
    loss_final_kernel<<<1, 256, 0, stream>>>(lossP, out);
}